// NystromAttention_72834055405757
// MI455X (gfx1250) — compile-verified
//
#include <hip/hip_runtime.h>

// ---------------------------------------------------------------------------
// Nystrom attention for MI455X (gfx1250, wave32).
// fp32 WMMA (v_wmma_f32_16x16x4_f32) for all large GEMMs; memory-bound design:
// Q/V each streamed once, K twice (k_bstats, k_bt_kv), small operands (Q_l,K_l,A_inv,kv,M) stay
// resident in the 192MB L2.
// Round-3 changes: reciprocal softmax sums (no v_div_scale chains in hot
// loops) and 4-way parallel softmax in the F/context kernel with float4
// F_t stores.
// ---------------------------------------------------------------------------

typedef __attribute__((ext_vector_type(2))) float v2f;
typedef __attribute__((ext_vector_type(8))) float v8f;

#define B_   4
#define H_   16
#define S_   4096
#define D_   128
#define L_   64
#define BH_  (B_ * H_)
#define EPS_ 1e-6f
#define SCALE_ 0.08838834764831845f   // 1/sqrt(128)

// D = A(16x4) * B(4x16) + C ; fp32, full precision
__device__ __forceinline__ v8f wmma_f32(v2f a, v2f b, v8f c) {
  return __builtin_amdgcn_wmma_f32_16x16x4_f32(false, a, false, b, (short)0, c,
                                               false, false);
}
__device__ __forceinline__ v2f ld2(const float* p) { return *(const v2f*)p; }

// ---------------------------------------------------------------------------
// K0: gather landmark rows Q_l, K_l   (indices are int64)
// ---------------------------------------------------------------------------
__global__ void k_gather(const float* __restrict__ Q, const float* __restrict__ K,
                         const long long* __restrict__ idx,
                         float* __restrict__ Qlw, float* __restrict__ Klw) {
  int i  = blockIdx.x * 256 + threadIdx.x;   // 0 .. BH*L*D-1 (524288)
  int d  = i & (D_ - 1);
  int l  = (i >> 7) & (L_ - 1);
  int bh = i >> 13;
  long long s = idx[l];
  size_t src = ((size_t)bh * S_ + (size_t)s) * D_ + d;
  Qlw[i] = Q[src];
  Klw[i] = K[src];
}

// ---------------------------------------------------------------------------
// K1: A_t = softmax(scale * Q_l @ K_l^T)  (64x64 per head) -> output A region
// ---------------------------------------------------------------------------
__global__ void k_At(const float* __restrict__ Qlw, const float* __restrict__ Klw,
                     float* __restrict__ Aout) {
  __shared__ float As[L_][L_ + 1];
  int bh = blockIdx.x;
  const float* Qb = Qlw + (size_t)bh * L_ * D_;
  const float* Kb = Klw + (size_t)bh * L_ * D_;
  for (int i = threadIdx.x; i < L_ * L_; i += blockDim.x) {
    int r = i >> 6, c = i & 63;
    const float4* q = (const float4*)(Qb + r * D_);
    const float4* k = (const float4*)(Kb + c * D_);
    float acc = 0.f;
    #pragma unroll 8
    for (int d = 0; d < D_ / 4; ++d) {
      float4 qv = q[d], kv = k[d];
      acc += qv.x * kv.x + qv.y * kv.y + qv.z * kv.z + qv.w * kv.w;
    }
    As[r][c] = acc * SCALE_;
  }
  __syncthreads();
  if (threadIdx.x < L_) {
    int r = threadIdx.x;
    float m = -3.4e38f;
    for (int c = 0; c < L_; ++c) m = fmaxf(m, As[r][c]);
    float s = 0.f;
    for (int c = 0; c < L_; ++c) { float e = __expf(As[r][c] - m); As[r][c] = e; s += e; }
    float inv = 1.f / s;
    float* o = Aout + ((size_t)bh * L_ + r) * L_;
    for (int c = 0; c < L_; ++c) o[c] = As[r][c] * inv;
  }
}

// ---------------------------------------------------------------------------
// K2: A_inv = inv(A_t + EPS*I)   Gauss-Jordan in LDS, 64 threads per head
// ---------------------------------------------------------------------------
__global__ void k_inv(const float* __restrict__ Ain, float* __restrict__ Ainv) {
  __shared__ float Aa[L_ * L_];
  __shared__ float Ii[L_ * L_];
  __shared__ float fcol[L_];
  int bh = blockIdx.x;
  int j  = threadIdx.x;                 // 64 threads, thread owns column j
  const float* src = Ain + (size_t)bh * L_ * L_;
  for (int i = j; i < L_ * L_; i += L_) {
    int r = i >> 6, c = i & 63;
    Aa[i] = src[i] + ((r == c) ? EPS_ : 0.f);
    Ii[i] = (r == c) ? 1.f : 0.f;
  }
  __syncthreads();
  for (int k = 0; k < L_; ++k) {
    float pinv = 1.0f / Aa[k * L_ + k];
    __syncthreads();
    Aa[k * L_ + j] *= pinv;
    Ii[k * L_ + j] *= pinv;
    __syncthreads();
    fcol[j] = Aa[j * L_ + k];           // capture factor column before writes
    __syncthreads();
    float akj = Aa[k * L_ + j], ikj = Ii[k * L_ + j];
    for (int i = 0; i < L_; ++i) {
      if (i == k) continue;
      float f = fcol[i];
      Aa[i * L_ + j] -= f * akj;
      Ii[i * L_ + j] -= f * ikj;
    }
    __syncthreads();
  }
  float* dst = Ainv + (size_t)bh * L_ * L_;
  for (int i = j; i < L_ * L_; i += L_) dst[i] = Ii[i];
}

// ---------------------------------------------------------------------------
// K3: per-row softmax stats of B_t logits over S: (max, 1/sum-exp)
//     one block per (bh, l), streaming online reduction
// ---------------------------------------------------------------------------
__global__ void k_bstats(const float* __restrict__ Qlw, const float* __restrict__ K,
                         float* __restrict__ stats) {
  __shared__ float qs[D_];
  __shared__ float rm[256], rs[256];
  int bh = blockIdx.x >> 6;
  int l  = blockIdx.x & 63;
  if (threadIdx.x < D_) qs[threadIdx.x] = Qlw[((size_t)bh * L_ + l) * D_ + threadIdx.x];
  __syncthreads();
  const float4* q4 = (const float4*)qs;
  float m = -3.4e38f, sum = 0.f;
  for (int s = threadIdx.x; s < S_; s += 256) {
    const float4* kp = (const float4*)(K + ((size_t)bh * S_ + s) * D_);
    float acc = 0.f;
    #pragma unroll 8
    for (int d = 0; d < D_ / 4; ++d) {
      float4 qv = q4[d], kv = kp[d];
      acc += qv.x * kv.x + qv.y * kv.y + qv.z * kv.z + qv.w * kv.w;
    }
    acc *= SCALE_;
    float nm = fmaxf(m, acc);
    sum = sum * __expf(m - nm) + __expf(acc - nm);
    m = nm;
  }
  rm[threadIdx.x] = m; rs[threadIdx.x] = sum;
  __syncthreads();
  for (int off = 128; off > 0; off >>= 1) {
    if (threadIdx.x < off) {
      float m1 = rm[threadIdx.x], s1 = rs[threadIdx.x];
      float m2 = rm[threadIdx.x + off], s2 = rs[threadIdx.x + off];
      float nm = fmaxf(m1, m2);
      rm[threadIdx.x] = nm;
      rs[threadIdx.x] = s1 * __expf(m1 - nm) + s2 * __expf(m2 - nm);
    }
    __syncthreads();
  }
  if (threadIdx.x == 0) {
    stats[(size_t)blockIdx.x * 2]     = rm[0];
    stats[(size_t)blockIdx.x * 2 + 1] = 1.0f / rs[0];   // reciprocal: mul in hot loop
  }
}

// ---------------------------------------------------------------------------
// K4: B_t (normalized, written to output) and kv = B_t @ V via WMMA.
//     One block per head; loops S in 64-wide tiles; kv accumulators persist
//     in registers (4 x v8f per wave).  Tile mapping shares the A fragment
//     across a wave's tiles (2x / 4x) for ILP and fewer loads.
// ---------------------------------------------------------------------------
__global__ void __launch_bounds__(256)
k_bt_kv(const float* __restrict__ Qlw, const float* __restrict__ K,
        const float* __restrict__ V, const float* __restrict__ stats,
        float* __restrict__ Bout, float* __restrict__ kvw) {
  __shared__ float Bs[L_][L_ + 1];
  __shared__ float smax[L_], srcp[L_];
  int bh   = blockIdx.x;
  int tid  = threadIdx.x;
  int w    = tid >> 5, lane = tid & 31;
  int half = lane >> 4, ln = lane & 15;
  if (tid < L_) {
    smax[tid] = stats[((size_t)bh * L_ + tid) * 2];
    srcp[tid] = stats[((size_t)bh * L_ + tid) * 2 + 1];
  }
  __syncthreads();

  // tile mapping
  const int tmL = w >> 1;                 // logits: row-tile (shared A)
  const int tnL = (w & 1) * 2;            // logits: first of 2 col-tiles
  const int tmK = w >> 1;                 // kv: row-tile (shared A)
  const int tnK = (w & 1) * 4;            // kv: first of 4 col-tiles

  v8f ckv[4];
  for (int t = 0; t < 4; ++t) for (int v = 0; v < 8; ++v) ckv[t][v] = 0.f;

  const float* Qb = Qlw + (size_t)bh * L_ * D_;
  const float* Kb = K   + (size_t)bh * S_ * D_;
  const float* Vb = V   + (size_t)bh * S_ * D_;

  const float* ap = Qb + (tmL * 16 + ln) * D_ + 2 * half;   // loop-invariant

  for (int st = 0; st < S_ / L_; ++st) {
    int s0 = st * L_;
    // ---- logits tiles: one A fragment -> two accumulators -----------------
    {
      v8f c0 = {}, c1 = {};
      const float* bp0 = Kb + (size_t)(s0 + tnL * 16 + ln) * D_ + 2 * half;
      const float* bp1 = bp0 + 16 * D_;
      #pragma unroll 4
      for (int kk = 0; kk < D_; kk += 4) {
        v2f a = ld2(ap + kk);
        c0 = wmma_f32(a, ld2(bp0 + kk), c0);
        c1 = wmma_f32(a, ld2(bp1 + kk), c1);
      }
      #pragma unroll
      for (int v = 0; v < 8; ++v) {
        int l = tmL * 16 + v + 8 * half;
        float mx = smax[l], rs = srcp[l];
        int sc0 = tnL * 16 + ln;
        float v0 = __expf(c0[v] * SCALE_ - mx) * rs;
        float v1 = __expf(c1[v] * SCALE_ - mx) * rs;
        Bs[l][sc0]      = v0;
        Bs[l][sc0 + 16] = v1;
        float* bo = Bout + ((size_t)bh * L_ + l) * S_ + s0 + sc0;
        bo[0]  = v0;
        bo[16] = v1;
      }
    }
    __syncthreads();
    // ---- kv += Bs(64x64) @ V_tile(64x128): shared A, 4 accumulators -------
    {
      int arow = tmK * 16 + ln;
      #pragma unroll 2
      for (int kk = 0; kk < L_; kk += 4) {
        int k0 = kk + 2 * half;
        v2f a; a[0] = Bs[arow][k0]; a[1] = Bs[arow][k0 + 1];
        const float* vr0 = Vb + (size_t)(s0 + k0) * D_;
        const float* vr1 = vr0 + D_;
        #pragma unroll
        for (int tt = 0; tt < 4; ++tt) {
          int bcol = (tnK + tt) * 16 + ln;
          v2f b; b[0] = vr0[bcol]; b[1] = vr1[bcol];
          ckv[tt] = wmma_f32(a, b, ckv[tt]);
        }
      }
    }
    __syncthreads();
  }
  for (int tt = 0; tt < 4; ++tt) {
    int tm = tmK, tn = tnK + tt;
    for (int v = 0; v < 8; ++v) {
      int l = tm * 16 + v + 8 * half;
      int d = tn * 16 + ln;
      kvw[(size_t)bh * L_ * D_ + l * D_ + d] = ckv[tt][v];
    }
  }
}

// ---------------------------------------------------------------------------
// K5: M = A_inv @ kv   (64x64 @ 64x128 per head -- tiny)
// ---------------------------------------------------------------------------
__global__ void k_m(const float* __restrict__ Ainv, const float* __restrict__ kvw,
                    float* __restrict__ Mw) {
  int bh = blockIdx.x;
  const float* Ab = Ainv + (size_t)bh * L_ * L_;
  const float* kb = kvw  + (size_t)bh * L_ * D_;
  float* Mb = Mw + (size_t)bh * L_ * D_;
  for (int i = threadIdx.x; i < L_ * D_; i += blockDim.x) {
    int r = i >> 7, d = i & 127;
    float acc = 0.f;
    #pragma unroll 8
    for (int l = 0; l < L_; ++l) acc += Ab[r * L_ + l] * kb[l * D_ + d];
    Mb[i] = acc;
  }
}

// ---------------------------------------------------------------------------
// K6: fused F_t + context:  F = softmax(scale * Q @ K_l^T) (written out),
//     context = F @ M.  One block per (head, 64-row S tile).
//     Softmax is 4-way parallel per row (256 thr = 64 rows x 4 quarters).
// ---------------------------------------------------------------------------
__global__ void __launch_bounds__(256)
k_f_ctx(const float* __restrict__ Q, const float* __restrict__ Klw,
        const float* __restrict__ Mw,
        float* __restrict__ Fout, float* __restrict__ Ctx) {
  __shared__ float Ls[L_][L_ + 1];
  __shared__ float red[L_][4];
  int bh = blockIdx.x >> 6;
  int st = blockIdx.x & 63;
  int s0 = st * 64;
  int tid = threadIdx.x, w = tid >> 5, lane = tid & 31;
  int half = lane >> 4, ln = lane & 15;
  const float* Qb = Q   + ((size_t)bh * S_ + s0) * D_;
  const float* Kb = Klw + (size_t)bh * L_ * D_;

  // ---- logits: one A fragment -> two accumulators -------------------------
  {
    int tm = w >> 1, tn0 = (w & 1) * 2;
    v8f c0 = {}, c1 = {};
    const float* ap  = Qb + (tm * 16 + ln) * D_ + 2 * half;
    const float* bp0 = Kb + (tn0 * 16 + ln) * D_ + 2 * half;
    const float* bp1 = bp0 + 16 * D_;
    #pragma unroll 4
    for (int kk = 0; kk < D_; kk += 4) {
      v2f a = ld2(ap + kk);
      c0 = wmma_f32(a, ld2(bp0 + kk), c0);
      c1 = wmma_f32(a, ld2(bp1 + kk), c1);
    }
    #pragma unroll
    for (int v = 0; v < 8; ++v) {
      int r = tm * 16 + v + 8 * half;
      Ls[r][tn0 * 16 + ln]      = c0[v] * SCALE_;
      Ls[r][tn0 * 16 + ln + 16] = c1[v] * SCALE_;
    }
  }
  __syncthreads();
  // ---- row softmax, 4 threads per row; write F_t with float4 stores -------
  {
    int r = tid >> 2, q = tid & 3;
    int c0 = q * 16;
    float m = -3.4e38f;
    #pragma unroll
    for (int c = 0; c < 16; ++c) m = fmaxf(m, Ls[r][c0 + c]);
    red[r][q] = m;
    __syncthreads();
    float mx = fmaxf(fmaxf(red[r][0], red[r][1]), fmaxf(red[r][2], red[r][3]));
    __syncthreads();
    float s = 0.f;
    float e[16];
    #pragma unroll
    for (int c = 0; c < 16; ++c) { e[c] = __expf(Ls[r][c0 + c] - mx); s += e[c]; }
    red[r][q] = s;
    __syncthreads();
    float inv = 1.0f / (red[r][0] + red[r][1] + red[r][2] + red[r][3]);
    float4* fo = (float4*)(Fout + ((size_t)bh * S_ + s0 + r) * L_ + c0);
    #pragma unroll
    for (int c4 = 0; c4 < 4; ++c4) {
      float4 fv;
      fv.x = e[c4 * 4 + 0] * inv;
      fv.y = e[c4 * 4 + 1] * inv;
      fv.z = e[c4 * 4 + 2] * inv;
      fv.w = e[c4 * 4 + 3] * inv;
      Ls[r][c0 + c4 * 4 + 0] = fv.x;
      Ls[r][c0 + c4 * 4 + 1] = fv.y;
      Ls[r][c0 + c4 * 4 + 2] = fv.z;
      Ls[r][c0 + c4 * 4 + 3] = fv.w;
      fo[c4] = fv;
    }
  }
  __syncthreads();
  // ---- context tile = F(64x64) @ M(64x128): shared A, 4 accumulators ------
  {
    const float* Mb = Mw + (size_t)bh * L_ * D_;
    int tm = w >> 1, tn0 = (w & 1) * 4;
    int arow = tm * 16 + ln;
    v8f c[4];
    for (int t = 0; t < 4; ++t) for (int v = 0; v < 8; ++v) c[t][v] = 0.f;
    #pragma unroll 2
    for (int kk = 0; kk < L_; kk += 4) {
      int k0 = kk + 2 * half;
      v2f a; a[0] = Ls[arow][k0]; a[1] = Ls[arow][k0 + 1];
      const float* mr0 = Mb + k0 * D_;
      const float* mr1 = mr0 + D_;
      #pragma unroll
      for (int tt = 0; tt < 4; ++tt) {
        int bcol = (tn0 + tt) * 16 + ln;
        v2f b; b[0] = mr0[bcol]; b[1] = mr1[bcol];
        c[tt] = wmma_f32(a, b, c[tt]);
      }
    }
    for (int tt = 0; tt < 4; ++tt)
      for (int v = 0; v < 8; ++v)
        Ctx[((size_t)bh * S_ + s0 + tm * 16 + v + 8 * half) * D_ + (tn0 + tt) * 16 + ln] =
            c[tt][v];
  }
}

// ---------------------------------------------------------------------------
extern "C" void kernel_launch(void* const* d_in, const int* in_sizes, int n_in,
                              void* d_out, int out_size, void* d_ws, size_t ws_size,
                              hipStream_t stream) {
  (void)in_sizes; (void)n_in; (void)out_size; (void)ws_size;
  const float*     Q   = (const float*)d_in[0];
  const float*     K   = (const float*)d_in[1];
  const float*     V   = (const float*)d_in[2];
  const long long* idx = (const long long*)d_in[3];

  float* out  = (float*)d_out;
  float* Ctx  = out;                       // B*H*S*D = 33554432
  float* Fout = out + 33554432;            // B*H*S*L = 16777216
  float* Aout = out + 50331648;            // B*H*L*L = 262144
  float* Bout = out + 50593792;            // B*H*L*S = 16777216

  float* ws    = (float*)d_ws;             // ~9.5 MB total
  float* Qlw   = ws;                       // 524288
  float* Klw   = ws + 524288;              // 524288
  float* Ainv  = ws + 1048576;             // 262144
  float* kvw   = ws + 1310720;             // 524288
  float* Mw    = ws + 1835008;             // 524288
  float* stats = ws + 2359296;             // 8192

  k_gather <<<2048, 256, 0, stream>>>(Q, K, idx, Qlw, Klw);
  k_At     <<<BH_,  256, 0, stream>>>(Qlw, Klw, Aout);
  k_inv    <<<BH_,  64,  0, stream>>>(Aout, Ainv);
  k_bstats <<<BH_ * L_, 256, 0, stream>>>(Qlw, K, stats);
  k_bt_kv  <<<BH_,  256, 0, stream>>>(Qlw, K, V, stats, Bout, kvw);
  k_m      <<<BH_,  256, 0, stream>>>(Ainv, kvw, Mw);
  k_f_ctx  <<<BH_ * L_, 256, 0, stream>>>(Q, Klw, Mw, Fout, Ctx);
}